// MyLightGCN_31885837206039
// MI455X (gfx1250) — compile-verified
//
#include <hip/hip_runtime.h>

#define N_USER   100000
#define N_ITEM   20000
#define N_NODES  120000
#define N_EDGES  2000000
#define EMB      64
#define XELEMS   (N_NODES * EMB)   // 7,680,000 floats

typedef float v2f __attribute__((ext_vector_type(2)));
typedef float v8f __attribute__((ext_vector_type(8)));

// ---------------------------------------------------------------------------
// CDNA5 async global->LDS gather (GVS mode: SGPR base + 32-bit VGPR offset).
// Tracked by ASYNCcnt; completion via s_wait_asynccnt.
// ---------------------------------------------------------------------------
__device__ __forceinline__ void async_gather_b128(unsigned lds_byte_addr,
                                                  const float* sbase,
                                                  unsigned goff_bytes) {
    asm volatile("global_load_async_to_lds_b128 %0, %1, %2"
                 :: "v"(lds_byte_addr), "v"(goff_bytes), "s"(sbase)
                 : "memory");
}
__device__ __forceinline__ void wait_asynccnt0() {
    asm volatile("s_wait_asynccnt 0" ::: "memory");
}

// ---------------------------------------------------------------------------
// 1) Build x = concat(user_emb, mean(item_emb, tag, testid, bigcat)) and
//    acc = alpha[0] * x.
// ---------------------------------------------------------------------------
__global__ void lg_build_x(const float* __restrict__ user_emb,
                           const float* __restrict__ item_emb,
                           const int*   __restrict__ item_tag,
                           const int*   __restrict__ item_testid,
                           const int*   __restrict__ item_bigcat,
                           const float* __restrict__ tag_emb,
                           const float* __restrict__ testid_emb,
                           const float* __restrict__ bigcat_emb,
                           const float* __restrict__ alpha,
                           float* __restrict__ x,
                           float* __restrict__ acc) {
    int idx = blockIdx.x * blockDim.x + threadIdx.x;
    if (idx >= XELEMS) return;
    int n = idx >> 6;     // node
    int d = idx & 63;     // dim
    float v;
    if (n < N_USER) {
        v = user_emb[idx];
    } else {
        int it = n - N_USER;
        v = 0.25f * (item_emb[it * EMB + d]
                   + tag_emb[item_tag[it] * EMB + d]
                   + testid_emb[item_testid[it] * EMB + d]
                   + bigcat_emb[item_bigcat[it] * EMB + d]);
    }
    x[idx]   = v;
    acc[idx] = alpha[0] * v;
}

// ---------------------------------------------------------------------------
// 2) deg[col[e]] += 1  (segment_sum of ones)
// ---------------------------------------------------------------------------
__global__ void lg_degree(const int* __restrict__ ecol,
                          float* __restrict__ deg) {
    int e = blockIdx.x * blockDim.x + threadIdx.x;
    if (e >= N_EDGES) return;
    atomicAdd(&deg[ecol[e]], 1.0f);
}

// ---------------------------------------------------------------------------
// 3) deg_inv_sqrt = deg > 0 ? rsqrt(max(deg,1)) : 0
// ---------------------------------------------------------------------------
__global__ void lg_deg_inv_sqrt(const float* __restrict__ deg,
                                float* __restrict__ dis) {
    int n = blockIdx.x * blockDim.x + threadIdx.x;
    if (n >= N_NODES) return;
    float dg = deg[n];
    dis[n] = (dg > 0.0f) ? rsqrtf(fmaxf(dg, 1.0f)) : 0.0f;
}

// ---------------------------------------------------------------------------
// 3b) Precompute per-edge norm = dis[row]*dis[col] (reused by both layers)
// ---------------------------------------------------------------------------
__global__ void lg_edge_norm(const int*   __restrict__ erow,
                             const int*   __restrict__ ecol,
                             const float* __restrict__ dis,
                             float*       __restrict__ enorm) {
    int e = blockIdx.x * blockDim.x + threadIdx.x;
    if (e >= N_EDGES) return;
    enorm[e] = dis[erow[e]] * dis[ecol[e]];
}

// ---------------------------------------------------------------------------
// 4) One wave (32 lanes) per edge: msg = x[row]*norm scatter-added into
//    xout[col]. Each lane owns a float2 slice of the 64-dim row.
// ---------------------------------------------------------------------------
__global__ void lg_propagate(const int*   __restrict__ erow,
                             const int*   __restrict__ ecol,
                             const float* __restrict__ enorm,
                             const float* __restrict__ xin,
                             float*       __restrict__ xout) {
    int t    = blockIdx.x * blockDim.x + threadIdx.x;
    int e    = t >> 5;
    int lane = t & 31;
    if (e >= N_EDGES) return;

    // speculative prefetch of upcoming edge data -> global_prefetch_b8
    __builtin_prefetch(erow + e + 2048, 0, 0);
    __builtin_prefetch(ecol + e + 2048, 0, 0);

    int   r    = erow[e];        // same address across the wave -> broadcast
    int   c    = ecol[e];
    float norm = enorm[e];

    float2 v = ((const float2*)xin)[r * 32 + lane];
    atomicAdd(&xout[c * EMB + lane * 2 + 0], v.x * norm);
    atomicAdd(&xout[c * EMB + lane * 2 + 1], v.y * norm);
}

// ---------------------------------------------------------------------------
// 5) acc += alpha[k] * xnew
// ---------------------------------------------------------------------------
__global__ void lg_axpy(float* __restrict__ acc,
                        const float* __restrict__ xnew,
                        const float* __restrict__ alpha,
                        int k) {
    int idx = blockIdx.x * blockDim.x + threadIdx.x;
    if (idx >= XELEMS) return;
    acc[idx] += alpha[k] * xnew[idx];
}

// ---------------------------------------------------------------------------
// 6) Batched edge dot products via V_WMMA_F32_16X16X4_F32 (exact fp32 math).
//    One wave handles 16 edges:
//      S = acc[src rows] (16 x 64), D = acc[dst rows] (16 x 64)
//      C = S * D^T accumulated over K in chunks of 4; result = diag(C).
//    Gather is done with GLOBAL_LOAD_ASYNC_TO_LDS_B128 (per-lane global
//    address -> per-lane LDS address, ASYNCcnt-tracked, wave-private wait).
//    A operand (16x4 f32): lanes 0-15 M=lane K={k0,k0+1},
//                          lanes 16-31 M=lane-16 K={k0+2,k0+3}.
//    B operand (4x16 f32): lanes 0-15 N=lane K={k0,k0+1},
//                          lanes 16-31 N=lane-16 K={k0+2,k0+3}.
//    With B[k][n] = D[n][k], both operands use the SAME LDS access pattern.
//    C layout: VGPR v -> rows M=v (lanes 0-15) and M=v+8 (lanes 16-31).
// ---------------------------------------------------------------------------
#define LDS_STRIDE 68           // 64 + 4 float pad: conflict-free, 16B-aligned rows
#define WAVES_PER_BLOCK 4

__global__ void lg_edge_dot_wmma(const int*   __restrict__ erow,
                                 const int*   __restrict__ ecol,
                                 const float* __restrict__ acc,
                                 float*       __restrict__ out) {
    __shared__ __align__(16) float lds[WAVES_PER_BLOCK * 2 * 16 * LDS_STRIDE];

    int lane = threadIdx.x & 31;
    int wave = threadIdx.x >> 5;
    int wave_global = blockIdx.x * WAVES_PER_BLOCK + wave;
    int e_base = wave_global * 16;

    float* A = &lds[wave * 2 * 16 * LDS_STRIDE];              // src rows
    float* B = A + 16 * LDS_STRIDE;                           // dst rows

    // Async cooperative gather: 16 rows x 16 float4 chunks per matrix.
    // 8 iterations/lane, 2 async B128 loads each -> ASYNCcnt peaks at 16.
    for (int i = lane; i < 256; i += 32) {
        int r  = i >> 4;          // local edge 0..15
        int c4 = i & 15;          // float4 column 0..15
        int e  = e_base + r;
        if (e >= N_EDGES) e = N_EDGES - 1;                    // clamp (no-op here)
        int s = erow[e];
        int d = ecol[e];
        unsigned ldsA = (unsigned)(uintptr_t)&A[r * LDS_STRIDE + c4 * 4];
        unsigned ldsB = (unsigned)(uintptr_t)&B[r * LDS_STRIDE + c4 * 4];
        unsigned gofS = (unsigned)(s * 256 + c4 * 16);        // byte offsets
        unsigned gofD = (unsigned)(d * 256 + c4 * 16);
        async_gather_b128(ldsA, acc, gofS);
        async_gather_b128(ldsB, acc, gofD);
    }
    wait_asynccnt0();   // wave-private: only this wave reads its LDS region

    int m    = lane & 15;
    int koff = (lane < 16) ? 0 : 2;
    const float* ar = &A[m * LDS_STRIDE + koff];
    const float* br = &B[m * LDS_STRIDE + koff];

    v8f c = {};
    for (int k0 = 0; k0 < EMB; k0 += 4) {
        v2f a, b;
        a.x = ar[k0];     a.y = ar[k0 + 1];
        b.x = br[k0];     b.y = br[k0 + 1];
        c = __builtin_amdgcn_wmma_f32_16x16x4_f32(
                /*neg_a=*/false, a, /*neg_b=*/false, b,
                /*c_mod=*/(short)0, c, /*reuse_a=*/false, /*reuse_b=*/false);
    }

    // Diagonal extraction without dynamic vector indexing (v_cndmask tree):
    // element (i,i) lives in VGPR i%8 at lane i (i<8) or lane i+16 (i>=8).
    union { v8f v; float f[8]; } cu;
    cu.v = c;
    int j = lane & 7;
    float s0 = (j & 4) ? cu.f[4] : cu.f[0];
    float s1 = (j & 4) ? cu.f[5] : cu.f[1];
    float s2 = (j & 4) ? cu.f[6] : cu.f[2];
    float s3 = (j & 4) ? cu.f[7] : cu.f[3];
    float t0 = (j & 2) ? s2 : s0;
    float t1 = (j & 2) ? s3 : s1;
    float val = (j & 1) ? t1 : t0;

    int i = -1;
    if (lane < 8)        i = lane;
    else if (lane >= 24) i = lane - 16;
    if (i >= 0) {
        int e = e_base + i;
        if (e < N_EDGES) out[e] = val;
    }
}

// ---------------------------------------------------------------------------
// Host launcher
// ---------------------------------------------------------------------------
extern "C" void kernel_launch(void* const* d_in, const int* in_sizes, int n_in,
                              void* d_out, int out_size, void* d_ws, size_t ws_size,
                              hipStream_t stream) {
    const int*   edge        = (const int*)d_in[0];
    const int*   erow        = edge;
    const int*   ecol        = edge + N_EDGES;
    const int*   item_tag    = (const int*)d_in[1];
    const int*   item_testid = (const int*)d_in[2];
    const int*   item_bigcat = (const int*)d_in[3];
    const float* user_emb    = (const float*)d_in[4];
    const float* item_emb    = (const float*)d_in[5];
    const float* tag_emb     = (const float*)d_in[6];
    const float* testid_emb  = (const float*)d_in[7];
    const float* bigcat_emb  = (const float*)d_in[8];
    const float* alpha       = (const float*)d_in[9];
    float*       out         = (float*)d_out;

    // Workspace carve-out (~101 MB)
    float* xA    = (float*)d_ws;
    float* xB    = xA    + XELEMS;
    float* acc   = xB    + XELEMS;
    float* deg   = acc   + XELEMS;
    float* dis   = deg   + N_NODES;
    float* enorm = dis   + N_NODES;

    hipMemsetAsync(deg, 0, N_NODES * sizeof(float), stream);
    hipMemsetAsync(xB,  0, XELEMS  * sizeof(float), stream);

    // Build x (into xA) and acc = alpha0 * x
    lg_build_x<<<(XELEMS + 255) / 256, 256, 0, stream>>>(
        user_emb, item_emb, item_tag, item_testid, item_bigcat,
        tag_emb, testid_emb, bigcat_emb, alpha, xA, acc);

    // Degrees, normalization, per-edge norm
    lg_degree<<<(N_EDGES + 255) / 256, 256, 0, stream>>>(ecol, deg);
    lg_deg_inv_sqrt<<<(N_NODES + 255) / 256, 256, 0, stream>>>(deg, dis);
    lg_edge_norm<<<(N_EDGES + 255) / 256, 256, 0, stream>>>(erow, ecol, dis, enorm);

    // Layer 1: xA -> xB ; acc += alpha1 * xB
    {
        long long threads = (long long)N_EDGES * 32;
        int blocks = (int)((threads + 255) / 256);
        lg_propagate<<<blocks, 256, 0, stream>>>(erow, ecol, enorm, xA, xB);
    }
    lg_axpy<<<(XELEMS + 255) / 256, 256, 0, stream>>>(acc, xB, alpha, 1);

    // Layer 2: xB -> xA ; acc += alpha2 * xA
    hipMemsetAsync(xA, 0, XELEMS * sizeof(float), stream);
    {
        long long threads = (long long)N_EDGES * 32;
        int blocks = (int)((threads + 255) / 256);
        lg_propagate<<<blocks, 256, 0, stream>>>(erow, ecol, enorm, xB, xA);
    }
    lg_axpy<<<(XELEMS + 255) / 256, 256, 0, stream>>>(acc, xA, alpha, 2);

    // Edge dot products: 16 edges per wave via fp32 WMMA + async LDS gather
    {
        int waves  = (N_EDGES + 15) / 16;                       // 125,000
        int blocks = (waves + WAVES_PER_BLOCK - 1) / WAVES_PER_BLOCK;
        lg_edge_dot_wmma<<<blocks, WAVES_PER_BLOCK * 32, 0, stream>>>(
            erow, ecol, acc, out);
    }
}